// SoftPadicRankingLoss_12335146074758
// MI455X (gfx1250) — compile-verified
//
#include <hip/hip_runtime.h>
#include <hip/hip_bf16.h>

#define N_ROWS 4096
#define DIMS   512
#define TILE   16
#define NTILES (N_ROWS / TILE)          // 256 row tiles
#define KSTEPS (DIMS / 4)               // 128 WMMA k-steps of 4
#define NWAVES 4
#define NSPLIT 4                        // column-space splits per row tile
#define BLK    (NWAVES * 32)
#define TSCALE (-2.0f)                  // -1/T, T = 0.5

typedef __attribute__((ext_vector_type(2))) float v2f;
typedef __attribute__((ext_vector_type(8))) float v8f;

// ---------------------------------------------------------------------------
// Kernel 1: row squared-norms. One wave per row, coalesced, shfl reduce.
// ---------------------------------------------------------------------------
__global__ __launch_bounds__(256) void sq_kernel(const float* __restrict__ z,
                                                 float* __restrict__ sq) {
  const int wave = threadIdx.x >> 5;
  const int lane = threadIdx.x & 31;
  const int row  = blockIdx.x * 8 + wave;
  const float* zr = z + (size_t)row * DIMS;
  float acc = 0.0f;
  #pragma unroll 4
  for (int i = lane; i < DIMS; i += 32) {
    float v = zr[i];
    acc += v * v;
  }
  #pragma unroll
  for (int m = 16; m >= 1; m >>= 1) acc += __shfl_xor(acc, m, 32);
  if (lane == 0) sq[row] = acc;
}

// ---------------------------------------------------------------------------
// Kernel 2: fused  z@z^T (WMMA f32 16x16x4) + latent exp-sums + 3-adic
// exp-sums.  Grid = NTILES * NSPLIT workgroups; each covers one 16-row tile
// and 1/NSPLIT of the column tiles, writing per-row (Zq, Zp, A) partials.
// ---------------------------------------------------------------------------
__global__ __launch_bounds__(BLK) void fused_kernel(
    const float* __restrict__ z, const int* __restrict__ idx,
    const float* __restrict__ sq, float* __restrict__ partials) {
  // A tile, swizzled so each wave's A fragment is a conflict-free ds_load_b64
  __shared__ float ldsA[KSTEPS * 64];          // 32 KB
  __shared__ float p3inv[16];                  // 3^-v
  __shared__ float et3[16];                    // exp(-2 * 3^-v)
  __shared__ float wsum[NWAVES][TILE][3];      // per-wave row partials

  const int tid     = threadIdx.x;
  const int wave    = tid >> 5;
  const int lane    = tid & 31;
  const int half    = lane >> 4;               // K-group selector
  const int l16     = lane & 15;
  const int rowTile = blockIdx.x & (NTILES - 1);
  const int split   = blockIdx.x >> 8;         // NTILES == 256
  const int rowBase = rowTile * TILE;

  if (tid < 16) {
    float p = 1.0f;
    for (int i = 0; i < tid; ++i) p *= (1.0f / 3.0f);
    p3inv[tid] = p;
    et3[tid]   = __expf(TSCALE * p);
  }

  // Stage A tile: element (m,k) -> dword kstep*64 + lane*2 + slot,
  // lane = 16*((k&3)>>1) + m, slot = k&1.  Global reads coalesced in k.
  for (int e = tid; e < TILE * DIMS; e += BLK) {
    const int m = e >> 9;
    const int k = e & (DIMS - 1);
    const int dw = ((k >> 2) << 6) + (((k >> 1) & 1) << 5) + (m << 1) + (k & 1);
    ldsA[dw] = z[(size_t)(rowBase + m) * DIMS + k];
  }
  __syncthreads();

  // Per-lane row constants (rows r + 8*half, r = 0..7)
  float sqRow[8];
  int   idxRow[8];
  #pragma unroll
  for (int r = 0; r < 8; ++r) {
    const int mg = rowBase + r + 8 * half;
    sqRow[r]  = sq[mg];
    idxRow[r] = idx[mg];
  }

  float zqAcc[8], zpAcc[8], aAcc[8];
  #pragma unroll
  for (int r = 0; r < 8; ++r) { zqAcc[r] = 0.0f; zpAcc[r] = 0.0f; aAcc[r] = 0.0f; }

  const v2f* aFrag = (const v2f*)&ldsA[lane * 2];   // stride 32 v2f per k-step

  // Epilogue for one 16x16 tile of the Gram matrix
  auto epi = [&](const v8f& g, int colBase) {
    const int   idxCol = idx[colBase + l16];
    const float sqCol  = sq[colBase + l16];
    #pragma unroll
    for (int r = 0; r < 8; ++r) {
      float d2 = sqRow[r] + sqCol - 2.0f * g[r];
      d2 = fmaxf(d2, 0.0f);
      const float s  = TSCALE * __builtin_sqrtf(d2);
      const float es = __expf(s);
      int diff = idxRow[r] - idxCol;
      diff = diff < 0 ? -diff : diff;
      float t, et;
      if (diff == 0) {
        t = 0.0f; et = 1.0f;
      } else {
        unsigned dd = (unsigned)diff;
        int v = 0;
        #pragma unroll
        for (int it = 0; it < 13; ++it) {           // 3^13 > MAX_IDX
          const unsigned q = dd / 3u;
          const bool div = (q * 3u == dd);
          v  = div ? v + 1 : v;
          dd = div ? q : dd;
        }
        t  = TSCALE * p3inv[v];
        et = et3[v];
      }
      zqAcc[r] += es;
      zpAcc[r] += et;
      aAcc[r]  += et * (t - s);
    }
  };

  // This wave handles column-tile pairs p = split*NWAVES + wave (mod 16)
  const int pStep = NWAVES * NSPLIT;
  for (int p = split * NWAVES + wave; p < NTILES / 2; p += pStep) {
    const int cb0 = (2 * p) * TILE;
    const int cb1 = cb0 + TILE;
    const float* zb0 = z + (size_t)(cb0 + l16) * DIMS + 2 * half;
    const float* zb1 = z + (size_t)(cb1 + l16) * DIMS + 2 * half;

    // Warm L2/L0 for the next pair this wave will touch
    const int pn = p + pStep;
    if (pn < NTILES / 2) {
      __builtin_prefetch(z + (size_t)(2 * pn * TILE + l16) * DIMS + 2 * half, 0, 0);
      __builtin_prefetch(z + (size_t)((2 * pn + 1) * TILE + l16) * DIMS + 2 * half, 0, 0);
    }

    v8f c0e, c0o, c1e, c1o;
    #pragma unroll
    for (int i = 0; i < 8; ++i) { c0e[i] = 0.f; c0o[i] = 0.f; c1e[i] = 0.f; c1o[i] = 0.f; }

    // Software-pipelined k-loop: loads for group k+2 in flight during group k
    v2f a0c  = aFrag[0];
    v2f a1c  = aFrag[32];
    v2f b00c = *(const v2f*)(zb0);
    v2f b01c = *(const v2f*)(zb0 + 4);
    v2f b10c = *(const v2f*)(zb1);
    v2f b11c = *(const v2f*)(zb1 + 4);

    for (int ks = 0; ks < KSTEPS - 2; ks += 2) {
      const v2f a0n  = aFrag[(ks + 2) * 32];
      const v2f a1n  = aFrag[(ks + 3) * 32];
      const v2f b00n = *(const v2f*)(zb0 + 4 * (ks + 2));
      const v2f b01n = *(const v2f*)(zb0 + 4 * (ks + 3));
      const v2f b10n = *(const v2f*)(zb1 + 4 * (ks + 2));
      const v2f b11n = *(const v2f*)(zb1 + 4 * (ks + 3));

      c0e = __builtin_amdgcn_wmma_f32_16x16x4_f32(false, a0c, false, b00c,
                                                  (short)0, c0e, false, false);
      c1e = __builtin_amdgcn_wmma_f32_16x16x4_f32(false, a0c, false, b10c,
                                                  (short)0, c1e, false, false);
      c0o = __builtin_amdgcn_wmma_f32_16x16x4_f32(false, a1c, false, b01c,
                                                  (short)0, c0o, false, false);
      c1o = __builtin_amdgcn_wmma_f32_16x16x4_f32(false, a1c, false, b11c,
                                                  (short)0, c1o, false, false);
      a0c = a0n; a1c = a1n;
      b00c = b00n; b01c = b01n; b10c = b10n; b11c = b11n;
    }
    // drain the last group (k-steps 126,127)
    c0e = __builtin_amdgcn_wmma_f32_16x16x4_f32(false, a0c, false, b00c,
                                                (short)0, c0e, false, false);
    c1e = __builtin_amdgcn_wmma_f32_16x16x4_f32(false, a0c, false, b10c,
                                                (short)0, c1e, false, false);
    c0o = __builtin_amdgcn_wmma_f32_16x16x4_f32(false, a1c, false, b01c,
                                                (short)0, c0o, false, false);
    c1o = __builtin_amdgcn_wmma_f32_16x16x4_f32(false, a1c, false, b11c,
                                                (short)0, c1o, false, false);

    epi(c0e + c0o, cb0);
    epi(c1e + c1o, cb1);
  }

  // Reduce across the 16 lanes of each half-wave (rows live per-half)
  #pragma unroll
  for (int r = 0; r < 8; ++r) {
    #pragma unroll
    for (int m = 1; m < 16; m <<= 1) {
      zqAcc[r] += __shfl_xor(zqAcc[r], m, 32);
      zpAcc[r] += __shfl_xor(zpAcc[r], m, 32);
      aAcc[r]  += __shfl_xor(aAcc[r], m, 32);
    }
  }
  if (l16 == 0) {
    #pragma unroll
    for (int r = 0; r < 8; ++r) {
      const int row = r + 8 * half;
      wsum[wave][row][0] = zqAcc[r];
      wsum[wave][row][1] = zpAcc[r];
      wsum[wave][row][2] = aAcc[r];
    }
  }
  __syncthreads();

  if (tid < TILE) {
    float zq = 0.f, zp = 0.f, aa = 0.f;
    for (int w = 0; w < NWAVES; ++w) {
      zq += wsum[w][tid][0];
      zp += wsum[w][tid][1];
      aa += wsum[w][tid][2];
    }
    float* p = partials + ((size_t)split * N_ROWS + rowBase + tid) * 3;
    p[0] = zq; p[1] = zp; p[2] = aa;
  }
}

// ---------------------------------------------------------------------------
// Kernel 3: deterministic finalize — combine split partials per row,
// apply the log/KL finish, tree-reduce 4096 rows to a scalar.
// ---------------------------------------------------------------------------
__global__ __launch_bounds__(1024) void finalize_kernel(
    const float* __restrict__ partials, float* __restrict__ out) {
  __shared__ float buf[1024];
  const int t = threadIdx.x;
  float acc = 0.0f;
  for (int row = t; row < N_ROWS; row += 1024) {
    float zq = 0.f, zp = 0.f, aa = 0.f;
    #pragma unroll
    for (int s = 0; s < NSPLIT; ++s) {
      const float* p = partials + ((size_t)s * N_ROWS + row) * 3;
      zq += p[0]; zp += p[1]; aa += p[2];
    }
    acc += aa / zp - __logf(zp) + __logf(zq);
  }
  buf[t] = acc;
  __syncthreads();
  for (int s = 512; s > 0; s >>= 1) {
    if (t < s) buf[t] += buf[t + s];
    __syncthreads();
  }
  if (t == 0) out[0] = buf[0] / (float)N_ROWS;
}

// ---------------------------------------------------------------------------
extern "C" void kernel_launch(void* const* d_in, const int* in_sizes, int n_in,
                              void* d_out, int out_size, void* d_ws,
                              size_t ws_size, hipStream_t stream) {
  const float* z   = (const float*)d_in[0];
  const int*   idx = (const int*)d_in[1];   // JAX default: int32 on device
  float* sq        = (float*)d_ws;          // N_ROWS floats
  float* partials  = sq + N_ROWS;           // NSPLIT * N_ROWS * 3 floats
  float* out       = (float*)d_out;

  sq_kernel<<<N_ROWS / 8, 256, 0, stream>>>(z, sq);
  fused_kernel<<<NTILES * NSPLIT, BLK, 0, stream>>>(z, idx, sq, partials);
  finalize_kernel<<<1, 1024, 0, stream>>>(partials, out);
}